// MDA_28656021799310
// MI455X (gfx1250) — compile-verified
//
#include <hip/hip_runtime.h>
#include <hip/hip_bf16.h>

// ---------------------------------------------------------------------------
// MDA fused pipeline for gfx1250 (CDNA5, wave32)
//  - WMMA f32 16x16x4 GEMMs, 64x64 block tile, 2 accumulators/wave
//  - double-buffered LDS staging via GLOBAL_LOAD_ASYNC_TO_LDS_B32 with
//    partial s_wait_asynccnt thresholds (in-order ASYNCcnt completion)
//  - K-pair-interleaved B tile so every WMMA operand is one ds_load_b64
// ---------------------------------------------------------------------------

typedef float v2f __attribute__((ext_vector_type(2)));
typedef float v8f __attribute__((ext_vector_type(8)));

#define BN_EPS 1e-5f

template <int Nw>
__device__ __forceinline__ void async_wait()
{
#if __has_builtin(__builtin_amdgcn_s_wait_asynccnt)
    __builtin_amdgcn_s_wait_asynccnt(Nw);
#else
    asm volatile("s_wait_asynccnt %0" :: "i"(Nw) : "memory");
#endif
}

// One per-lane async 4B copy global -> LDS (wave-relative LDS address =
// low 32 bits of the generic pointer, per CDNA5 aperture mapping).
__device__ __forceinline__ void async_g2l_b32(const float* gp, const float* lp)
{
    unsigned lds = (unsigned)(size_t)lp;
    asm volatile("global_load_async_to_lds_b32 %0, %1, off"
                 :: "v"(lds), "v"(gp) : "memory");
}

// ------------------------- WMMA GEMM -----------------------------------
// C[M,N] = A[M,K] @ op(B)  (+bias[N]) (+relu) (+row-slice remap)
//   transB=1: B stored [N,K] row-major -> A @ B^T
//   transB=0: B stored [K,N] row-major -> A @ B
// slice_start>0: rows r<901 -> r ; r>=slice_start -> r-slice_start+901 ; else dropped.
#define TM 64
#define TN 64
#define TK 32

// Stage one K-strip into the given LDS buffer. Returns the number of async
// instructions this wave issued (0 / 8 / 16) -> used as s_wait_asynccnt bound.
__device__ __forceinline__ int stage_tile(
    const float* __restrict__ A, const float* __restrict__ B,
    float (*As_)[TK + 2], float (*Bs_)[TN][2],
    int t, int tileM, int tileN, int k0,
    int M, int K, int N, int lda, int ldb, int transB,
    bool fullM, bool fullN)
{
    const bool fullK = (k0 + TK <= K);
    int cnt = 0;

    // ---- A tile [TM x TK], row-major, rows padded to 34 floats ----
    if (fullM && fullK) {
        #pragma unroll
        for (int i = 0; i < (TM * TK) / 256; ++i) {       // 8 per thread
            int l = t + i * 256;
            int r = l >> 5, c = l & 31;
            async_g2l_b32(A + (long)(tileM + r) * lda + (k0 + c), &As_[r][c]);
        }
        cnt += 8;
    } else {
        #pragma unroll
        for (int i = 0; i < (TM * TK) / 256; ++i) {
            int l = t + i * 256;
            int r = l >> 5, c = l & 31;
            int gr = tileM + r, gc = k0 + c;
            As_[r][c] = (gr < M && gc < K) ? A[(long)gr * lda + gc] : 0.0f;
        }
    }
    // ---- B tile, K-pair-interleaved: Bs_[k>>1][n][k&1] = B[k][n] ----
    if (fullN && fullK) {
        #pragma unroll
        for (int i = 0; i < (TK * TN) / 256; ++i) {       // 8 per thread
            int l = t + i * 256;
            int kk = l >> 6, nn = l & 63;
            const float* gp = transB ? (B + (long)(tileN + nn) * ldb + (k0 + kk))
                                     : (B + (long)(k0 + kk) * ldb + (tileN + nn));
            async_g2l_b32(gp, &Bs_[kk >> 1][nn][kk & 1]);
        }
        cnt += 8;
    } else {
        #pragma unroll
        for (int i = 0; i < (TK * TN) / 256; ++i) {
            int l = t + i * 256;
            int kk = l >> 6, nn = l & 63;
            int gk = k0 + kk, gn = tileN + nn;
            float v = 0.0f;
            if (gk < K && gn < N)
                v = transB ? B[(long)gn * ldb + gk] : B[(long)gk * ldb + gn];
            Bs_[kk >> 1][nn][kk & 1] = v;
        }
    }
    return cnt;
}

__global__ __launch_bounds__(256)
void gemm_wmma_f32(const float* __restrict__ A, const float* __restrict__ B,
                   float* __restrict__ C, const float* __restrict__ bias,
                   int M, int K, int N, int lda, int ldb, int ldc,
                   int transB, int relu, int slice_start)
{
    __shared__ float As[2][TM][TK + 2];       // 8B-aligned (k,k+1) pairs
    __shared__ float Bs[2][TK / 2][TN][2];    // pair-interleaved

    const int t     = threadIdx.x;
    const int tileM = blockIdx.y * TM;
    const int tileN = blockIdx.x * TN;
    const int wave  = t >> 5;        // 8 waves of 32
    const int lane  = t & 31;
    const int wm    = wave & 3;      // 4 wave-tiles along M (16 rows each)
    const int wn    = wave >> 2;     // 2 wave-tiles along N (32 cols each)
    const int ml    = lane & 15;
    const int hi    = lane >> 4;     // lane half -> K pair {0,1} vs {2,3}

    v8f acc[2] = {};

    const bool fullM = (tileM + TM <= M);
    const bool fullN = (tileN + TN <= N);
    const int  ktiles = (K + TK - 1) / TK;

    // prologue: prefetch K-strip 0 into buffer 0
    stage_tile(A, B, As[0], Bs[0], t, tileM, tileN, 0,
               M, K, N, lda, ldb, transB, fullM, fullN);

    for (int kt = 0; kt < ktiles; ++kt) {
        const int cur = kt & 1;
        int cnt = 0;
        if (kt + 1 < ktiles)           // prefetch next strip into other buffer
            cnt = stage_tile(A, B, As[cur ^ 1], Bs[cur ^ 1], t, tileM, tileN,
                             (kt + 1) * TK, M, K, N, lda, ldb, transB, fullM, fullN);
        // Async loads complete in order: leaving <=cnt outstanding guarantees
        // the *current* strip has fully landed while the next keeps streaming.
        if (cnt == 16)     async_wait<16>();
        else if (cnt == 8) async_wait<8>();
        else               async_wait<0>();
        __syncthreads();

        #pragma unroll
        for (int kf = 0; kf < TK; kf += 4) {
            // f32 16x16x4 fragments (ISA 7.12.2):
            //  A: lanes0-15 K={kf,kf+1}, lanes16-31 K={kf+2,kf+3}
            //  B: VGPR0 rows kf/kf+2 ; VGPR1 rows kf+1/kf+3  (pair = 1x b64)
            v2f a  = *(const v2f*)&As[cur][wm * 16 + ml][kf + 2 * hi];
            v2f b0 = *(const v2f*)&Bs[cur][(kf >> 1) + hi][wn * 32 + ml][0];
            v2f b1 = *(const v2f*)&Bs[cur][(kf >> 1) + hi][wn * 32 + 16 + ml][0];
            acc[0] = __builtin_amdgcn_wmma_f32_16x16x4_f32(
                false, a, false, b0, (short)0, acc[0], false, false);
            acc[1] = __builtin_amdgcn_wmma_f32_16x16x4_f32(
                false, a, false, b1, (short)0, acc[1], false, false);
        }
        __syncthreads();
    }

    // --- epilogue: bias + relu + slice remap store ---
    #pragma unroll
    for (int sub = 0; sub < 2; ++sub) {
        int gn = tileN + wn * 32 + sub * 16 + ml;
        if (gn >= N) continue;
        float bvv = bias ? bias[gn] : 0.0f;
        #pragma unroll
        for (int r = 0; r < 8; ++r) {
            int gm = tileM + wm * 16 + r + hi * 8;   // C: VGPR r -> rows r / r+8
            if (gm >= M) continue;
            float v = acc[sub][r] + bvv;
            if (relu) v = v > 0.0f ? v : 0.0f;
            int orow = gm;
            if (slice_start > 0) {
                if (gm < 901)               orow = gm;
                else if (gm >= slice_start) orow = gm - slice_start + 901;
                else                        continue;
            }
            C[(long)orow * ldc + gn] = v;
        }
    }
}

// --------------------- GAT attention helpers ----------------------------
__global__ __launch_bounds__(128)
void rowdot2(const float* __restrict__ h, const float* __restrict__ a1,
             const float* __restrict__ a2, float* __restrict__ o1,
             float* __restrict__ o2, int Kd)
{
    __shared__ float s1[128], s2[128];
    const long row = blockIdx.x;
    float d1 = 0.0f, d2 = 0.0f;
    for (int k = threadIdx.x; k < Kd; k += 128) {
        float v = h[row * Kd + k];
        d1 += v * a1[k];
        d2 += v * a2[k];
    }
    s1[threadIdx.x] = d1; s2[threadIdx.x] = d2;
    __syncthreads();
    for (int s = 64; s > 0; s >>= 1) {
        if (threadIdx.x < s) {
            s1[threadIdx.x] += s1[threadIdx.x + s];
            s2[threadIdx.x] += s2[threadIdx.x + s];
        }
        __syncthreads();
    }
    if (threadIdx.x == 0) { o1[row] = s1[0]; o2[row] = s2[0]; }
}

// e[i,j] = mask ? leaky_relu(adst[i]+asrc[j], 0.2) : -1e30,
// mask = (adj[j,i]!=0) || (i==j)   (LDS-tiled transpose of adj)
__global__ __launch_bounds__(256)
void build_e(const int* __restrict__ adj, const float* __restrict__ asrc,
             const float* __restrict__ adst, float* __restrict__ e, int N)
{
    __shared__ int tile[32][33];
    const int i0 = blockIdx.x * 32;
    const int j0 = blockIdx.y * 32;
    const int tx = threadIdx.x & 31;
    const int ty = threadIdx.x >> 5;
    for (int yy = ty; yy < 32; yy += 8) {
        int j = j0 + yy, i = i0 + tx;
        tile[yy][tx] = (j < N && i < N) ? adj[(long)j * N + i] : 0;
    }
    __syncthreads();
    for (int yy = ty; yy < 32; yy += 8) {
        int i = i0 + yy, j = j0 + tx;
        if (i < N && j < N) {
            float s  = adst[i] + asrc[j];
            float lr = s > 0.0f ? s : 0.2f * s;
            bool  m  = (tile[tx][yy] != 0) || (i == j);
            e[(long)i * N + j] = m ? lr : -1e30f;
        }
    }
}

__global__ __launch_bounds__(256)
void softmax_rows(float* __restrict__ e, int N)
{
    __shared__ float sm[256];
    const long base = (long)blockIdx.x * N;
    float mx = -3.4e38f;
    for (int j = threadIdx.x; j < N; j += 256) mx = fmaxf(mx, e[base + j]);
    sm[threadIdx.x] = mx; __syncthreads();
    for (int s = 128; s > 0; s >>= 1) {
        if (threadIdx.x < s) sm[threadIdx.x] = fmaxf(sm[threadIdx.x], sm[threadIdx.x + s]);
        __syncthreads();
    }
    mx = sm[0]; __syncthreads();
    float sum = 0.0f;
    for (int j = threadIdx.x; j < N; j += 256) sum += __expf(e[base + j] - mx);
    sm[threadIdx.x] = sum; __syncthreads();
    for (int s = 128; s > 0; s >>= 1) {
        if (threadIdx.x < s) sm[threadIdx.x] += sm[threadIdx.x + s];
        __syncthreads();
    }
    float inv = 1.0f / sm[0];
    for (int j = threadIdx.x; j < N; j += 256) e[base + j] = __expf(e[base + j] - mx) * inv;
}

// -------------------------- MS-CAM --------------------------------------
__global__ void cam_mix(const float* __restrict__ v, const float* __restrict__ md,
                        const float* __restrict__ lw1, const float* __restrict__ lb1,
                        const float* __restrict__ gw1, const float* __restrict__ gb1,
                        float* __restrict__ y1l, float* __restrict__ y1g, int npix)
{
    int idx = blockIdx.x * 256 + threadIdx.x;
    if (idx >= npix) return;
    float x0 = v[idx], x1 = v[npix + idx], x2 = v[2 * (long)npix + idx], x3 = md[idx];
    y1l[idx] = lw1[0] * x0 + lw1[1] * x1 + lw1[2] * x2 + lw1[3] * x3 + lb1[0];
    y1g[idx] = gw1[0] * x0 + gw1[1] * x1 + gw1[2] * x2 + gw1[3] * x3 + gb1[0];
}

__global__ __launch_bounds__(256)
void stats_partial(const float* __restrict__ a, const float* __restrict__ b,
                   float* __restrict__ part, int npix)
{
    __shared__ float sh[4][256];
    float sa = 0, qa = 0, sb = 0, qb = 0;
    for (long i = blockIdx.x * 256 + threadIdx.x; i < npix; i += (long)gridDim.x * 256) {
        float va = a[i], vb = b[i];
        sa += va; qa += va * va; sb += vb; qb += vb * vb;
    }
    sh[0][threadIdx.x] = sa; sh[1][threadIdx.x] = qa;
    sh[2][threadIdx.x] = sb; sh[3][threadIdx.x] = qb;
    __syncthreads();
    for (int s = 128; s > 0; s >>= 1) {
        if (threadIdx.x < s)
            for (int c = 0; c < 4; ++c) sh[c][threadIdx.x] += sh[c][threadIdx.x + s];
        __syncthreads();
    }
    if (threadIdx.x == 0)
        for (int c = 0; c < 4; ++c) part[blockIdx.x * 4 + c] = sh[c][0];
}

__global__ void stats_finalize(const float* __restrict__ part, float* __restrict__ stats,
                               int nblocks, int npix, int statoff)
{
    if (threadIdx.x != 0 || blockIdx.x != 0) return;
    float s0 = 0, s1 = 0, s2 = 0, s3 = 0;
    for (int i = 0; i < nblocks; ++i) {      // fixed order -> deterministic
        s0 += part[i * 4 + 0]; s1 += part[i * 4 + 1];
        s2 += part[i * 4 + 2]; s3 += part[i * 4 + 3];
    }
    float inv = 1.0f / (float)npix;
    float ma = s0 * inv, va = s1 * inv - ma * ma;
    float mb = s2 * inv, vb = s3 * inv - mb * mb;
    stats[statoff + 0] = ma; stats[statoff + 1] = va;
    stats[statoff + 2] = mb; stats[statoff + 3] = vb;
}

__global__ void cam_relunorm(float* __restrict__ y1l, float* __restrict__ y1g,
                             const float* __restrict__ stats, int npix)
{
    int idx = blockIdx.x * 256 + threadIdx.x;
    if (idx >= npix) return;
    float il = rsqrtf(stats[1] + BN_EPS), ig = rsqrtf(stats[3] + BN_EPS);
    float r1 = (y1l[idx] - stats[0]) * il;
    float r2 = (y1g[idx] - stats[2]) * ig;
    y1l[idx] = r1 > 0.0f ? r1 : 0.0f;
    y1g[idx] = r2 > 0.0f ? r2 : 0.0f;
}

// BN(w2*r + b2) collapses to w2*(r-mean_r)/sqrt(w2^2*var_r + eps) (bias cancels)
__global__ void cam_fuse(const float* __restrict__ v, const float* __restrict__ md,
                         const float* __restrict__ rl, const float* __restrict__ rg,
                         const float* __restrict__ lw2, const float* __restrict__ gw2,
                         const float* __restrict__ stats, float* __restrict__ sumx, int npix)
{
    int idx = blockIdx.x * 256 + threadIdx.x;
    if (idx >= npix) return;
    float mrl = stats[4], vrl = stats[5], mrg = stats[6], vrg = stats[7];
    float r_l = rl[idx] - mrl;
    float r_g = rg[idx] - mrg;
    float x[4] = { v[idx], v[npix + idx], v[2 * (long)npix + idx], md[idx] };
    float acc = 0.0f;
    #pragma unroll
    for (int c = 0; c < 4; ++c) {
        float wl = lw2[c], wg = gw2[c];
        float xl = wl * r_l * rsqrtf(wl * wl * vrl + BN_EPS);
        float xg = wg * r_g * rsqrtf(wg * wg * vrg + BN_EPS);
        float s  = 1.0f / (1.0f + __expf(-(xl + xg)));
        acc += x[c] * s;
    }
    sumx[idx] = acc * 0.25f;
}

// ----------------------------- MLP --------------------------------------
__global__ void gather_emb(const float* __restrict__ sumx, const int* __restrict__ ts,
                           float* __restrict__ emb, int npairs)
{
    int idx = blockIdx.x * 256 + threadIdx.x;
    int total = npairs * 1802;
    if (idx >= total) return;
    int p = idx / 1802, c = idx - p * 1802;
    int which = c >= 901;
    int row = ts[p * 2 + which];
    if (row < 0) row = 0; if (row >= 1778) row = 1777;
    int col = c - which * 901;
    emb[(long)p * 1802 + c] = sumx[(long)row * 901 + col];
}

__global__ void mlp_final(const float* __restrict__ h3, const float* __restrict__ w,
                          const float* __restrict__ b, float* __restrict__ out, int npairs)
{
    int p = blockIdx.x * 256 + threadIdx.x;
    if (p >= npairs) return;
    float acc = b[0];
    #pragma unroll
    for (int k = 0; k < 64; ++k) acc += h3[(long)p * 64 + k] * w[k];
    out[p] = acc;
}

// --------------------------- orchestration ------------------------------
static inline int cdiv(int a, int b) { return (a + b - 1) / b; }

extern "C" void kernel_launch(void* const* d_in, const int* in_sizes, int n_in,
                              void* d_out, int out_size, void* d_ws, size_t ws_size,
                              hipStream_t stream)
{
    (void)in_sizes; (void)n_in; (void)out_size; (void)ws_size;

    const int*   adjv[3]  = { (const int*)d_in[0], (const int*)d_in[1], (const int*)d_in[2] };
    const int*   ts       = (const int*)d_in[3];
    const float* featv[3] = { (const float*)d_in[4], (const float*)d_in[5], (const float*)d_in[6] };
    const float* md       = (const float*)d_in[7];
    const float* Wv[3]    = { (const float*)d_in[8],  (const float*)d_in[12], (const float*)d_in[16] };
    const float* asv[3]   = { (const float*)d_in[9],  (const float*)d_in[13], (const float*)d_in[17] };
    const float* adv[3]   = { (const float*)d_in[10], (const float*)d_in[14], (const float*)d_in[18] };
    const float* bv[3]    = { (const float*)d_in[11], (const float*)d_in[15], (const float*)d_in[19] };
    const float* lw1 = (const float*)d_in[20]; const float* lb1 = (const float*)d_in[21];
    const float* lw2 = (const float*)d_in[22];
    const float* gw1 = (const float*)d_in[24]; const float* gb1 = (const float*)d_in[25];
    const float* gw2 = (const float*)d_in[26];
    const float* mW1 = (const float*)d_in[28]; const float* mb1 = (const float*)d_in[29];
    const float* mW2 = (const float*)d_in[30]; const float* mb2 = (const float*)d_in[31];
    const float* mW3 = (const float*)d_in[32]; const float* mb3 = (const float*)d_in[33];
    const float* mW4 = (const float*)d_in[34]; const float* mb4 = (const float*)d_in[35];

    const int Nv[3]    = { 2060, 2459, 3929 };
    const int slice[3] = { 1183, 1582, 3052 };
    const int OUT   = 901;
    const int NROWS = 1778;
    const int NPIX  = NROWS * OUT;            // 1,601,978
    const int NP    = 4096;

    float* ws = (float*)d_ws;
    const long O_V    = 0;                    // 3 fused views [3][1778*901]
    const long O_SUMX = O_V + 3L * NPIX;      // fused sum
    const long O_H    = O_SUMX + NPIX;        // h / y1 maps / h2,h3
    const long SZ_H   = 3929L * 901;
    const long O_E    = O_H + SZ_H;           // e,attn / emb,h1
    const long SZ_E   = 3929L * 3929;
    const long O_AS   = O_E + SZ_E;
    const long O_STAT = O_AS + 8192;
    const long O_PART = O_STAT + 32;

    float* v_out = ws + O_V;
    float* sumx  = ws + O_SUMX;
    float* hbuf  = ws + O_H;
    float* ebuf  = ws + O_E;
    float* asrc  = ws + O_AS;
    float* adst  = ws + O_AS + 4096;
    float* stats = ws + O_STAT;
    float* part  = ws + O_PART;

    // ================= per-view GAT =================
    for (int v = 0; v < 3; ++v) {
        const int N = Nv[v];
        gemm_wmma_f32<<<dim3(cdiv(OUT, TN), cdiv(N, TM)), 256, 0, stream>>>(
            featv[v], Wv[v], hbuf, nullptr, N, N, OUT, N, N, OUT, 1, 0, 0);
        rowdot2<<<N, 128, 0, stream>>>(hbuf, asv[v], adv[v], asrc, adst, OUT);
        build_e<<<dim3(cdiv(N, 32), cdiv(N, 32)), 256, 0, stream>>>(adjv[v], asrc, adst, ebuf, N);
        softmax_rows<<<N, 256, 0, stream>>>(ebuf, N);
        gemm_wmma_f32<<<dim3(cdiv(OUT, TN), cdiv(N, TM)), 256, 0, stream>>>(
            ebuf, hbuf, v_out + (long)v * NPIX, bv[v], N, N, OUT, N, OUT, OUT, 0, 1, slice[v]);
    }

    // ================= MS-CAM fusion =================
    float* y1l = hbuf;
    float* y1g = hbuf + NPIX;
    const int pixBlocks = cdiv(NPIX, 256);
    cam_mix<<<pixBlocks, 256, 0, stream>>>(v_out, md, lw1, lb1, gw1, gb1, y1l, y1g, NPIX);
    stats_partial<<<512, 256, 0, stream>>>(y1l, y1g, part, NPIX);
    stats_finalize<<<1, 1, 0, stream>>>(part, stats, 512, NPIX, 0);
    cam_relunorm<<<pixBlocks, 256, 0, stream>>>(y1l, y1g, stats, NPIX);
    stats_partial<<<512, 256, 0, stream>>>(y1l, y1g, part, NPIX);
    stats_finalize<<<1, 1, 0, stream>>>(part, stats, 512, NPIX, 4);
    cam_fuse<<<pixBlocks, 256, 0, stream>>>(v_out, md, y1l, y1g, lw2, gw2, stats, sumx, NPIX);

    // ================= pair MLP =================
    float* emb = ebuf;                       // [4096,1802]
    float* h1  = ebuf + (long)NP * 1802;     // [4096,1024]
    float* h2  = hbuf;                       // [4096,512]
    float* h3  = hbuf + (long)NP * 512;      // [4096,64]

    gather_emb<<<cdiv(NP * 1802, 256), 256, 0, stream>>>(sumx, ts, emb, NP);
    gemm_wmma_f32<<<dim3(cdiv(1024, TN), cdiv(NP, TM)), 256, 0, stream>>>(
        emb, mW1, h1, mb1, NP, 1802, 1024, 1802, 1802, 1024, 1, 0, 0);
    gemm_wmma_f32<<<dim3(cdiv(512, TN), cdiv(NP, TM)), 256, 0, stream>>>(
        h1, mW2, h2, mb2, NP, 1024, 512, 1024, 1024, 512, 1, 0, 0);
    gemm_wmma_f32<<<dim3(cdiv(64, TN), cdiv(NP, TM)), 256, 0, stream>>>(
        h2, mW3, h3, mb3, NP, 512, 64, 512, 512, 64, 1, 0, 0);
    mlp_final<<<cdiv(NP, 256), 256, 0, stream>>>(h3, mW4, mb4, (float*)d_out, NP);
}